// GNN_3255585210493
// MI455X (gfx1250) — compile-verified
//
#include <hip/hip_runtime.h>
#include <hip/hip_bf16.h>

typedef __attribute__((ext_vector_type(2))) float v2f;
typedef __attribute__((ext_vector_type(8))) float v8f;

#define N_NODES 100000
#define IN_D    64
#define HID_D   128
#define OUT_D   64

// ----------------- degree / symmetric norm -----------------
__global__ void deg_init_k(float* __restrict__ deg, int n) {
  int i = blockIdx.x * blockDim.x + threadIdx.x;
  if (i < n) deg[i] = 1.0f;  // self loop contributes 1
}

__global__ void deg_count_k(const long long* __restrict__ dst, float* deg, int E) {
  int e = blockIdx.x * blockDim.x + threadIdx.x;
  if (e < E) atomicAdd(&deg[dst[e]], 1.0f);
}

__global__ void deg_rsqrt_k(float* __restrict__ deg, int n) {
  int i = blockIdx.x * blockDim.x + threadIdx.x;
  if (i < n) deg[i] = rsqrtf(deg[i]);  // deg >= 1 always
}

// ----------------- fp32 WMMA GEMM -----------------
// C[M x N] = A[M x K] @ B[K x N], row-major. One wave computes one 16x16 C
// tile with V_WMMA_F32_16X16X4_F32. blockDim.x = (N/16)*32, gridDim.x = M/16.
// M must be a multiple of 16 (100000 = 6250*16) so EXEC stays all-ones.
//
// Per CDNA5 ISA layouts (CDNA5 ISA):
//  A 16x4:  lanes 0-15 -> M=lane, VGPR{0,1}=K{0,1}; lanes 16-31 -> K{2,3}
//  B 4x16:  lanes 0-15 -> N=lane, VGPR{0,1}=K{0,1}; lanes 16-31 -> K{2,3}
//  C/D:     VGPR i: lanes 0-15 -> M=i, lanes 16-31 -> M=i+8, N=lane&15
template<int K, int N, bool RELU_A>
__global__ void gemm_wmma_k(const float* __restrict__ A, const float* __restrict__ B,
                            float* __restrict__ C) {
  const int wave = threadIdx.x >> 5;
  const int lane = threadIdx.x & 31;
  const int half = lane >> 4;     // K-pair select
  const int ml   = lane & 15;     // row of A / col of B
  const size_t mbase = (size_t)blockIdx.x * 16;
  const int nbase = wave * 16;

  const float* Arow = A + (mbase + ml) * K;
  v8f acc = {};
#pragma unroll
  for (int k = 0; k < K; k += 4) {
    const int ka = k + 2 * half;
    float a0 = Arow[ka];
    float a1 = Arow[ka + 1];
    if (RELU_A) { a0 = fmaxf(a0, 0.0f); a1 = fmaxf(a1, 0.0f); }
    v2f a, b;
    a.x = a0;
    a.y = a1;
    b.x = B[(size_t)ka * N + nbase + ml];
    b.y = B[(size_t)(ka + 1) * N + nbase + ml];
    // (neg_a, A, neg_b, B, c_mod, C, reuse_a, reuse_b)
    acc = __builtin_amdgcn_wmma_f32_16x16x4_f32(false, a, false, b,
                                                (short)0, acc, false, false);
  }
  float* Crow = C + (mbase + (size_t)half * 8) * N + nbase + ml;
#pragma unroll
  for (int i = 0; i < 8; ++i)
    Crow[(size_t)i * N] = acc[i];
}

// ----------------- aggregation -----------------
// agg[n][f] = hlin[n][f] * dinv[n]^2 + bias[f]   (self-loop term + bias)
template<int F>
__global__ void agg_init_k(const float* __restrict__ hlin, const float* __restrict__ dinv,
                           const float* __restrict__ bias, float* __restrict__ agg) {
  long long idx = (long long)blockIdx.x * blockDim.x + threadIdx.x;
  if (idx >= (long long)N_NODES * F) return;
  int node = (int)(idx / F);
  int f    = (int)(idx % F);
  float di = dinv[node];
  agg[idx] = hlin[idx] * di * di + bias[f];
}

// F/4 lanes per edge: coalesced float4 gather of hlin[src], scaled scatter-add
// into agg[dst] via global_atomic_add_f32 (lands in L2; agg buffer is L2-resident).
template<int F>
__global__ void agg_edge_k(const float* __restrict__ hlin, const float* __restrict__ dinv,
                           const long long* __restrict__ src, const long long* __restrict__ dst,
                           float* __restrict__ agg, int E) {
  constexpr int C = F / 4;
  long long idx = (long long)blockIdx.x * blockDim.x + threadIdx.x;
  long long e = idx / C;
  int c = (int)(idx % C);
  if (e >= E) return;
  long long s = src[e];
  long long d = dst[e];
  float norm = dinv[s] * dinv[d];
  const float4 v = *(const float4*)(hlin + s * (long long)F + c * 4);
  float* out = agg + d * (long long)F + c * 4;
  atomicAdd(out + 0, v.x * norm);
  atomicAdd(out + 1, v.y * norm);
  atomicAdd(out + 2, v.z * norm);
  atomicAdd(out + 3, v.w * norm);
}

// ----------------- edge classifier -----------------
// One wave32 per edge: lane l covers features {l, l+32} of src row and dst row.
// ReLU of layer-2 pre-activations applied here (h2 buffer holds pre-ReLU sums).
__global__ void edge_cls_k(const float* __restrict__ h2,
                           const long long* __restrict__ src,
                           const long long* __restrict__ dst,
                           const float* __restrict__ Wc, const float* __restrict__ bc,
                           float* __restrict__ out, int E) {
  int lane = threadIdx.x & 31;
  long long e = ((long long)blockIdx.x * blockDim.x + threadIdx.x) >> 5;
  if (e >= E) return;
  long long s = src[e];
  long long d = dst[e];
  float w0 = Wc[lane], w1 = Wc[lane + 32], w2 = Wc[lane + 64], w3 = Wc[lane + 96];
  float a0 = fmaxf(h2[s * OUT_D + lane],      0.0f);
  float a1 = fmaxf(h2[s * OUT_D + lane + 32], 0.0f);
  float a2 = fmaxf(h2[d * OUT_D + lane],      0.0f);
  float a3 = fmaxf(h2[d * OUT_D + lane + 32], 0.0f);
  float acc = a0 * w0 + a1 * w1 + a2 * w2 + a3 * w3;
#pragma unroll
  for (int off = 16; off > 0; off >>= 1)
    acc += __shfl_xor(acc, off, 32);
  if (lane == 0) {
    float z = acc + bc[0];
    out[e] = 1.0f / (1.0f + expf(-z));
  }
}

extern "C" void kernel_launch(void* const* d_in, const int* in_sizes, int n_in,
                              void* d_out, int out_size, void* d_ws, size_t ws_size,
                              hipStream_t stream) {
  const float*     x  = (const float*)d_in[0];
  const long long* ei = (const long long*)d_in[1];   // int64 edge_index [2, E]
  const float*     W1 = (const float*)d_in[2];
  const float*     b1 = (const float*)d_in[3];
  const float*     W2 = (const float*)d_in[4];
  const float*     b2 = (const float*)d_in[5];
  const float*     Wc = (const float*)d_in[6];
  const float*     bc = (const float*)d_in[7];
  float* out = (float*)d_out;

  const int E = in_sizes[1] / 2;
  const long long* src = ei;
  const long long* dst = ei + E;

  // workspace: dinv | bufA (100k x 128) | bufB (100k x 128)  ~= 103 MB
  float* ws   = (float*)d_ws;
  float* dinv = ws;
  float* bufA = ws + 100352;                         // padded past 100000
  float* bufB = bufA + (size_t)N_NODES * HID_D;

  const int TB = 256;

  // --- symmetric normalization coefficients ---
  deg_init_k<<<(N_NODES + TB - 1) / TB, TB, 0, stream>>>(dinv, N_NODES);
  deg_count_k<<<(E + TB - 1) / TB, TB, 0, stream>>>(dst, dinv, E);
  deg_rsqrt_k<<<(N_NODES + TB - 1) / TB, TB, 0, stream>>>(dinv, N_NODES);

  // --- layer 1: h1lin = x @ W1 (bufA), aggregate -> bufB (bias folded, pre-ReLU) ---
  gemm_wmma_k<IN_D, HID_D, false>
      <<<N_NODES / 16, (HID_D / 16) * 32, 0, stream>>>(x, W1, bufA);
  agg_init_k<HID_D>
      <<<((long long)N_NODES * HID_D + TB - 1) / TB, TB, 0, stream>>>(bufA, dinv, b1, bufB);
  agg_edge_k<HID_D>
      <<<((long long)E * (HID_D / 4) + TB - 1) / TB, TB, 0, stream>>>(bufA, dinv, src, dst, bufB, E);

  // --- layer 2: h2lin = relu(h1) @ W2 (bufA), aggregate -> bufB (pre-ReLU) ---
  gemm_wmma_k<HID_D, OUT_D, true>
      <<<N_NODES / 16, (OUT_D / 16) * 32, 0, stream>>>(bufB, W2, bufA);
  agg_init_k<OUT_D>
      <<<((long long)N_NODES * OUT_D + TB - 1) / TB, TB, 0, stream>>>(bufA, dinv, b2, bufB);
  agg_edge_k<OUT_D>
      <<<((long long)E * (OUT_D / 4) + TB - 1) / TB, TB, 0, stream>>>(bufA, dinv, src, dst, bufB, E);

  // --- edge scorer: sigmoid(concat(relu(h2[s]), relu(h2[d])) @ Wc + bc) ---
  edge_cls_k<<<((long long)E * 32 + TB - 1) / TB, TB, 0, stream>>>(bufB, src, dst, Wc, bc, out, E);
}